// Attn_45019847197338
// MI455X (gfx1250) — compile-verified
//
#include <hip/hip_runtime.h>
#include <math.h>

// Problem constants (match reference)
#define B_DIM 16
#define S_DIM 4096
#define H_DIM 1024

typedef float v2f __attribute__((ext_vector_type(2)));
typedef float v8f __attribute__((ext_vector_type(8)));

// ---------------------------------------------------------------------------
// Kernel 1: u = hidden(16x1024) @ W(1024x1024), fp32 WMMA 16x16x4.
// One wave (32 threads) per 16-wide N tile; K loop of 256 steps of 4.
//
// VGPR layouts per CDNA5 ISA 7.12.2:
//   A 16x4 f32 : lane L holds M=L%16; VGPR0=K{0|2}, VGPR1=K{1|3} (half-wave split)
//   B 4x16 f32 : lane L holds N=L%16; VGPR0=K{0|2}, VGPR1=K{1|3}
//   D 16x16 f32: lane L, VGPR v -> M = v + (L<16 ? 0 : 8), N = L%16
// ---------------------------------------------------------------------------
__global__ void hiddenW_wmma(const float* __restrict__ hidden,
                             const float* __restrict__ W,
                             float* __restrict__ u) {
    const int lane  = threadIdx.x;            // 0..31
    const int ntile = blockIdx.x;             // 0..63
    const int n     = (lane & 15);
    const int ncol  = ntile * 16 + n;
    const int mrow  = (lane & 15);
    const int khalf = (lane >> 4) * 2;        // 0 for lanes 0-15, 2 for 16-31

    v8f acc = {};
    for (int k = 0; k < H_DIM; k += 4) {
        v2f a, b;
        a.x = hidden[mrow * H_DIM + k + khalf];
        a.y = hidden[mrow * H_DIM + k + khalf + 1];
        b.x = W[(k + khalf)     * H_DIM + ncol];
        b.y = W[(k + khalf + 1) * H_DIM + ncol];
        // 8 args: (neg_a, A, neg_b, B, c_mod, C, reuse_a, reuse_b)
        acc = __builtin_amdgcn_wmma_f32_16x16x4_f32(
            false, a, false, b, (short)0, acc, false, false);
    }

    const int rbase = (lane >> 4) * 8;        // 0 or 8
#pragma unroll
    for (int v = 0; v < 8; ++v) {
        u[(rbase + v) * H_DIM + ncol] = acc[v];
    }
}

// ---------------------------------------------------------------------------
// Kernel 2: scores[b,s] = dot(enc[b,s,:], u[b,:])   (bandwidth-bound, 268 MB)
// Block = 256 threads (8 waves). Grid = (S/32, B). u[b] cached in LDS.
// Each wave reduces one full 1024-float row via float4 loads + shfl_xor.
// ---------------------------------------------------------------------------
__global__ void score_k(const float* __restrict__ enc,
                        const float* __restrict__ u,
                        float* __restrict__ scores) {
    __shared__ float su[H_DIM];
    const int b   = blockIdx.y;
    const int tid = threadIdx.x;              // 0..255

    for (int i = tid; i < H_DIM; i += 256)
        su[i] = u[b * H_DIM + i];
    __syncthreads();

    const int lane = tid & 31;
    const int wave = tid >> 5;                // 0..7
    const int s0   = blockIdx.x * 32;         // 32 rows per block
    const float4* u4 = (const float4*)su;

#pragma unroll
    for (int r = 0; r < 4; ++r) {
        const int s = s0 + wave + r * 8;
        const float4* e4 =
            (const float4*)(enc + ((size_t)b * S_DIM + s) * H_DIM);
        float acc = 0.0f;
#pragma unroll
        for (int j = 0; j < 8; ++j) {         // 8 * 32 lanes * 4 = 1024 elems
            float4 ev = e4[lane + j * 32];
            float4 uv = u4[lane + j * 32];
            acc += ev.x * uv.x + ev.y * uv.y + ev.z * uv.z + ev.w * uv.w;
        }
#pragma unroll
        for (int off = 16; off > 0; off >>= 1)
            acc += __shfl_xor(acc, off, 32);
        if (lane == 0)
            scores[b * S_DIM + s] = acc;
    }
}

// ---------------------------------------------------------------------------
// Kernel 3: out[b,s] = softmax over s of scores[b,:].
// One block (256 threads) per batch row; 16 values per thread.
// Bias term hidden[b]·bias is constant over s -> exactly cancelled by softmax.
// ---------------------------------------------------------------------------
__global__ void softmax_k(const float* __restrict__ scores,
                          float* __restrict__ out) {
    const int b    = blockIdx.x;
    const int tid  = threadIdx.x;             // 0..255
    const int lane = tid & 31;
    const int wave = tid >> 5;                // 0..7
    __shared__ float red[8];

    float vals[16];
    float m = -INFINITY;
#pragma unroll
    for (int i = 0; i < 16; ++i) {
        vals[i] = scores[b * S_DIM + tid + i * 256];
        m = fmaxf(m, vals[i]);
    }
#pragma unroll
    for (int off = 16; off > 0; off >>= 1)
        m = fmaxf(m, __shfl_xor(m, off, 32));
    if (lane == 0) red[wave] = m;
    __syncthreads();
    {
        float mm = red[0];
#pragma unroll
        for (int w = 1; w < 8; ++w) mm = fmaxf(mm, red[w]);
        m = mm;
    }
    __syncthreads();

    float sum = 0.0f;
#pragma unroll
    for (int i = 0; i < 16; ++i) {
        vals[i] = __expf(vals[i] - m);
        sum += vals[i];
    }
#pragma unroll
    for (int off = 16; off > 0; off >>= 1)
        sum += __shfl_xor(sum, off, 32);
    if (lane == 0) red[wave] = sum;
    __syncthreads();
    {
        float ss = 0.0f;
#pragma unroll
        for (int w = 0; w < 8; ++w) ss += red[w];
        sum = ss;
    }
    const float inv = 1.0f / sum;
#pragma unroll
    for (int i = 0; i < 16; ++i)
        out[b * S_DIM + tid + i * 256] = vals[i] * inv;
}

// ---------------------------------------------------------------------------
extern "C" void kernel_launch(void* const* d_in, const int* in_sizes, int n_in,
                              void* d_out, int out_size, void* d_ws, size_t ws_size,
                              hipStream_t stream) {
    (void)in_sizes; (void)n_in; (void)out_size; (void)ws_size;

    const float* hidden = (const float*)d_in[0];   // [16,1024]
    const float* enc    = (const float*)d_in[1];   // [16,4096,1024]
    const float* W      = (const float*)d_in[2];   // [1024,1024]
    // d_in[3] = bias[1024]: unused — constant-over-s term is removed exactly
    // by softmax shift invariance.

    float* out    = (float*)d_out;                 // [16,1,4096]
    float* u      = (float*)d_ws;                  // 16*1024 floats (64 KB)
    float* scores = u + B_DIM * H_DIM;             // 16*4096 floats (256 KB)

    hiddenW_wmma<<<dim3(H_DIM / 16), dim3(32), 0, stream>>>(hidden, W, u);
    score_k<<<dim3(S_DIM / 32, B_DIM), dim3(256), 0, stream>>>(enc, u, scores);
    softmax_k<<<dim3(B_DIM), dim3(256), 0, stream>>>(scores, out);
}